// MeanAggMPNN_uw_54485955117462
// MI455X (gfx1250) — compile-verified
//
#include <hip/hip_runtime.h>
#include <hip/hip_bf16.h>

// MPNN (scatter-mean + 2x 128x128 GEMM + 128x2 head) for MI455X / gfx1250.
// - scatter-mean: wave-per-edge, float4 gathers, fp32 HW atomics (L2-resident)
// - GEMMs: V_WMMA_F32_16X16X4_F32 (exact fp32), A/W staged in LDS
// - head: wave-per-node shfl reduction (wave32)

typedef __attribute__((ext_vector_type(2))) float v2f;
typedef __attribute__((ext_vector_type(8))) float v8f;

#define FDIM 128

// ---------------------------------------------------------------- degree cnt
__global__ __launch_bounds__(256) void count_kernel(const int* __restrict__ dst,
                                                    float* __restrict__ cnt,
                                                    int n_edges) {
    int e = blockIdx.x * blockDim.x + threadIdx.x;
    if (e < n_edges) unsafeAtomicAdd(&cnt[dst[e]], 1.0f);
}

__global__ __launch_bounds__(256) void inv_kernel(float* __restrict__ cnt, int n) {
    int i = blockIdx.x * blockDim.x + threadIdx.x;
    if (i < n) cnt[i] = 1.0f / fmaxf(cnt[i], 1.0f);
}

// ------------------------------------------------------------- scatter (sum)
// One wave (32 lanes) per edge; each lane moves one float4 (128 floats/edge).
__global__ __launch_bounds__(256) void scatter_add_kernel(
    const float* __restrict__ feat, const int* __restrict__ srcIdx,
    const int* __restrict__ dstIdx, float* __restrict__ out, int n_edges) {
    int wid  = (blockIdx.x * blockDim.x + threadIdx.x) >> 5;
    int lane = threadIdx.x & 31;
    if (wid >= n_edges) return;
    int s = srcIdx[wid];
    int d = dstIdx[wid];
    const float4 v = ((const float4*)(feat + (size_t)s * FDIM))[lane];
    float* fd = out + (size_t)d * FDIM + lane * 4;
    unsafeAtomicAdd(fd + 0, v.x);
    unsafeAtomicAdd(fd + 1, v.y);
    unsafeAtomicAdd(fd + 2, v.z);
    unsafeAtomicAdd(fd + 3, v.w);
}

// ----------------------------------------------- H = relu((A*inv) @ W + b)
// Block: 256 thr = 8 waves. Tile: 16 nodes x 128 outputs; wave w owns cols
// [16w,16w+16). K swept in 32-row LDS panels of W, 8 WMMA f32 16x16x4 each.
__global__ __launch_bounds__(256) void gemm_relu_wmma(
    const float* __restrict__ A, const float* __restrict__ inv,
    const float* __restrict__ W, const float* __restrict__ bias,
    float* __restrict__ H, int n_nodes) {
    __shared__ __align__(16) float As[16][FDIM + 4];  // +4 pad: conflict-free col reads
    __shared__ __align__(16) float Ws[32][FDIM];

    const int tid   = threadIdx.x;
    const int wave  = tid >> 5;
    const int lane  = tid & 31;
    const int node0 = blockIdx.x * 16;

    // Load A tile, pre-scaled by 1/deg (2048 floats, 8 per thread).
    {
        int base = tid * 8;
        int r = base >> 7;         // node row 0..15
        int c = base & 127;
        int row = node0 + r;
        float s = (row < n_nodes) ? inv[row] : 0.0f;
        const float* src = A + (size_t)row * FDIM + c;
        #pragma unroll
        for (int j = 0; j < 8; ++j)
            As[r][c + j] = (row < n_nodes) ? src[j] * s : 0.0f;
    }
    __syncthreads();

    v8f acc = {};
    const int m   = lane & 15;
    const int hi  = lane >> 4;         // half-wave select
    const int col = wave * 16 + m;

    for (int kp = 0; kp < FDIM; kp += 32) {
        // Stage 32x128 W panel into LDS (4096 floats, 4 x float4 per thread).
        {
            int base = tid * 16;
            int r = base >> 7;
            int c = base & 127;
            const float4* s4 = (const float4*)(W + (size_t)(kp + r) * FDIM + c);
            float4* d4 = (float4*)(&Ws[r][c]);
            #pragma unroll
            for (int j = 0; j < 4; ++j) d4[j] = s4[j];
            if (kp + 32 < FDIM)  // prefetch next panel -> global_prefetch
                __builtin_prefetch(W + (size_t)(kp + 32 + r) * FDIM + c, 0, 0);
        }
        __syncthreads();

        #pragma unroll
        for (int k0 = 0; k0 < 32; k0 += 4) {
            // A-frag 16x4 f32: v0 = {K=k0 | K=k0+2}, v1 = {K=k0+1 | K=k0+3}
            int ka = k0 + 2 * hi;
            v2f a, b;
            a.x = As[m][kp + ka];
            a.y = As[m][kp + ka + 1];
            // B-frag 4x16 f32 (mirrored row striping), N = col
            b.x = Ws[ka][col];
            b.y = Ws[ka + 1][col];
            acc = __builtin_amdgcn_wmma_f32_16x16x4_f32(
                false, a, false, b, (short)0, acc, false, false);
        }
        __syncthreads();
    }

    // C/D layout: VGPR r -> M=r (lanes 0-15) / M=8+r (lanes 16-31), N=lane&15
    float bv = bias[col];
    #pragma unroll
    for (int r = 0; r < 8; ++r) {
        int row = node0 + (hi ? 8 + r : r);
        if (row < n_nodes) {
            float v = acc[r] + bv;
            H[(size_t)row * FDIM + col] = v > 0.0f ? v : 0.0f;
        }
    }
}

// ------------------------------------------------------ out = H @ W3 + b3
// Wave per node; lane-parallel dot over 128, wave32 shfl tree reduction.
__global__ __launch_bounds__(256) void final_linear(
    const float* __restrict__ H, const float* __restrict__ W3,
    const float* __restrict__ b3, float* __restrict__ out, int n_nodes) {
    int wid  = (blockIdx.x * blockDim.x + threadIdx.x) >> 5;
    int lane = threadIdx.x & 31;
    if (wid >= n_nodes) return;
    const float* h = H + (size_t)wid * FDIM;
    float a0 = 0.0f, a1 = 0.0f;
    #pragma unroll
    for (int j = 0; j < 4; ++j) {
        int k = lane + 32 * j;
        float v = h[k];
        a0 = fmaf(v, W3[k * 2 + 0], a0);
        a1 = fmaf(v, W3[k * 2 + 1], a1);
    }
    #pragma unroll
    for (int off = 16; off > 0; off >>= 1) {
        a0 += __shfl_down(a0, off, 32);
        a1 += __shfl_down(a1, off, 32);
    }
    if (lane == 0) {
        out[(size_t)wid * 2 + 0] = a0 + b3[0];
        out[(size_t)wid * 2 + 1] = a1 + b3[1];
    }
}

// --------------------------------------------------------------------------
extern "C" void kernel_launch(void* const* d_in, const int* in_sizes, int n_in,
                              void* d_out, int out_size, void* d_ws, size_t ws_size,
                              hipStream_t stream) {
    const float* x     = (const float*)d_in[0];
    const int*   edges = (const int*)d_in[1];   // [2,E] (int32 per harness contract)
    const float* W1    = (const float*)d_in[2];
    const float* b1    = (const float*)d_in[3];
    const float* W2    = (const float*)d_in[4];
    const float* b2    = (const float*)d_in[5];
    const float* W3    = (const float*)d_in[6];
    const float* b3    = (const float*)d_in[7];
    float*       outp  = (float*)d_out;

    const int N = in_sizes[0] / FDIM;   // 20000
    const int E = in_sizes[1] / 2;      // 640000
    const int* src = edges;
    const int* dst = edges + E;

    // Workspace: cnt->inv [N], agg A [N*128], hidden H [N*128]  (~20.6 MB)
    float* cnt = (float*)d_ws;
    float* A   = cnt + ((N + 255) & ~255);
    float* H   = A + (size_t)N * FDIM;

    const int tpb = 256;
    const int edgeWaveBlocks = (E * 32 + tpb - 1) / tpb;
    const int nodeWaveBlocks = (N * 32 + tpb - 1) / tpb;
    const int gemmBlocks     = (N + 15) / 16;

    // degree and its reciprocal (shared by both layers)
    hipMemsetAsync(cnt, 0, sizeof(float) * (size_t)N, stream);
    count_kernel<<<(E + tpb - 1) / tpb, tpb, 0, stream>>>(dst, cnt, E);
    inv_kernel<<<(N + tpb - 1) / tpb, tpb, 0, stream>>>(cnt, N);

    // layer 1
    hipMemsetAsync(A, 0, sizeof(float) * (size_t)N * FDIM, stream);
    scatter_add_kernel<<<edgeWaveBlocks, tpb, 0, stream>>>(x, src, dst, A, E);
    gemm_relu_wmma<<<gemmBlocks, tpb, 0, stream>>>(A, cnt, W1, b1, H, N);

    // layer 2 (reuse A; H1 consumed by scatter, then overwritten with H2)
    hipMemsetAsync(A, 0, sizeof(float) * (size_t)N * FDIM, stream);
    scatter_add_kernel<<<edgeWaveBlocks, tpb, 0, stream>>>(H, src, dst, A, E);
    gemm_relu_wmma<<<gemmBlocks, tpb, 0, stream>>>(A, cnt, W2, b2, H, N);

    // head
    final_linear<<<nodeWaveBlocks, tpb, 0, stream>>>(H, W3, b3, outp, N);
}